// HardAttentionSoftmax_1623497638470
// MI455X (gfx1250) — compile-verified
//
#include <hip/hip_runtime.h>

// Task-conditional linear + binary hard-argmax for MI455X (gfx1250, wave32).
//
// Strategy: bucket samples by task (20 tasks), then run one skinny f32 GEMM
// per task with V_WMMA_F32_16X16X4_F32 (exact f32, matches reference math).
// Memory-bound: 80 MB of unique weights (fits in 192 MB L2), ~3.4 us at
// 23.3 TB/s HBM. WMMA turns the per-sample matvecs into shared-weight GEMMs.

typedef __attribute__((ext_vector_type(2))) float v2f;
typedef __attribute__((ext_vector_type(8))) float v8f;

#define FIN    1024
#define FOUT2  1024
#define BATCH  256
#define NTASKS 20

// ---------------------------------------------------------------------------
// Kernel 1: bucket samples by task, pad each bucket to a multiple of 16 with
// -1 sentinels, and zero a 1024-float scratch row (A-padding source).
// Single block, 256 threads (one per sample).
// ---------------------------------------------------------------------------
__global__ __launch_bounds__(256) void cl_setup(const int* __restrict__ task_id,
                                                float* __restrict__ zrow,
                                                int* __restrict__ counts,
                                                int* __restrict__ slots) {
  const int tid = threadIdx.x;
  // Zero the padding row (harness poisons workspace with 0xAA).
#pragma unroll
  for (int i = 0; i < FIN / 256; ++i) zrow[tid + i * 256] = 0.0f;
  if (tid < NTASKS) counts[tid] = 0;
  __syncthreads();
  const int t = task_id[tid];
  const int pos = atomicAdd(&counts[t], 1);
  slots[t * BATCH + pos] = tid;
  __syncthreads();
  if (tid < NTASKS) {
    const int c = counts[tid];
    const int cpad = (c + 15) & ~15;
    for (int i = c; i < cpad; ++i) slots[tid * BATCH + i] = -1;
  }
}

// ---------------------------------------------------------------------------
// Kernel 2: per-task GEMM  logits[s, n] = sum_k x[s, k] * W[t][n*FIN + k]
// Grid: (16 N-chunks of 64, 20 tasks). Block: 128 threads = 4 waves, each
// wave owns one 16-wide N-subtile and loops over 16-row M-tiles of its task.
//
// f32 WMMA 16x16x4 operand layouts (ISA 7.12.2):
//   A (16x4): lane l -> row (l&15), K-pair 2*(l>>4)   -> float2 load from x row
//   B (4x16): lane l -> col (l&15), K-pair 2*(l>>4)   -> float2 load from W col
//   C (16x16): VGPR r -> row r (lanes 0-15) / r+8 (lanes 16-31), col = lane&15
// ---------------------------------------------------------------------------
__global__ __launch_bounds__(128) void cl_gemm(const float* __restrict__ x,
                                               const float* __restrict__ W,
                                               const float* __restrict__ zrow,
                                               const int* __restrict__ counts,
                                               const int* __restrict__ slots,
                                               float* __restrict__ logits) {
  const int t = blockIdx.y;
  const int cnt = counts[t];           // block-uniform
  if (cnt == 0) return;

  const int lane  = threadIdx.x & 31;
  const int wave  = threadIdx.x >> 5;
  const int n0    = blockIdx.x * 64 + wave * 16;
  const int l15   = lane & 15;
  const int lhalf = lane >> 4;         // 0: K=0,1   1: K=2,3
  const int koff  = lhalf * 2;

  const float* Wt   = W + (size_t)t * ((size_t)FIN * FOUT2);
  const float* bcol = Wt + (size_t)(n0 + l15) * FIN + koff;   // B column, contiguous in K

  for (int m0 = 0; m0 < cnt; m0 += 16) {
    // Lane l15 holds the slot (sample index or -1) for tile row l15.
    const int sidx = slots[t * BATCH + m0 + l15];
    // Padded rows read from the zeroed scratch row -> EXEC stays all-1s,
    // no per-iteration branching in the K loop.
    const float* arow = ((sidx >= 0) ? (x + (size_t)sidx * FIN) : zrow) + koff;

    v8f acc = {};
#pragma unroll 16
    for (int k = 0; k < FIN; k += 4) {
      v2f af = *(const v2f*)(arow + k);
      v2f bf = *(const v2f*)(bcol + k);
      acc = __builtin_amdgcn_wmma_f32_16x16x4_f32(
          /*neg_a=*/false, af, /*neg_b=*/false, bf,
          /*c_mod=*/(short)0, acc, /*reuse_a=*/false, /*reuse_b=*/false);
    }

    // Scatter C back to each sample's logits row. The per-row slot is a
    // lane-permute of sidx (lane `row` holds slots[m0+row]) -> ds_bpermute,
    // no extra global loads in the epilogue.
#pragma unroll
    for (int r = 0; r < 8; ++r) {
      const int row = lhalf ? (r + 8) : r;
      const int s2 = __builtin_amdgcn_ds_bpermute(row << 2, sidx);
      if (s2 >= 0) logits[(size_t)s2 * FOUT2 + n0 + l15] = acc[r];
    }
  }
}

// ---------------------------------------------------------------------------
// Kernel 3: h[b, j] = 1 - argmax(logits[b, 2j], logits[b, 2j+1])
// jnp.argmax takes the FIRST max on ties -> h = (l1 > l0) ? 0 : 1.
// ---------------------------------------------------------------------------
__global__ __launch_bounds__(256) void cl_argmax(const float* __restrict__ logits,
                                                 float* __restrict__ h) {
  const int i = blockIdx.x * blockDim.x + threadIdx.x;   // 0 .. 131071
  const v2f l = *(const v2f*)(logits + 2 * (size_t)i);   // one b64 load per pair
  h[i] = (l.y > l.x) ? 0.0f : 1.0f;
}

// ---------------------------------------------------------------------------
extern "C" void kernel_launch(void* const* d_in, const int* in_sizes, int n_in,
                              void* d_out, int out_size, void* d_ws, size_t ws_size,
                              hipStream_t stream) {
  const float* x       = (const float*)d_in[0];   // [256, 1024]
  const int*   task_id = (const int*)d_in[1];     // [256]
  const float* W       = (const float*)d_in[2];   // [20, 1024*1024]

  float* out_h      = (float*)d_out;                       // [256, 512]
  float* out_logits = out_h + (size_t)BATCH * (FOUT2 / 2); // [256, 1024]

  // Workspace layout (~25 KB): zero row | counts(32) | slots[20][256]
  float* zrow   = (float*)d_ws;
  int*   counts = (int*)(zrow + FIN);
  int*   slots  = counts + 32;

  cl_setup<<<1, 256, 0, stream>>>(task_id, zrow, counts, slots);
  cl_gemm<<<dim3(FOUT2 / 64, NTASKS), 128, 0, stream>>>(x, W, zrow, counts, slots,
                                                        out_logits);
  cl_argmax<<<(BATCH * (FOUT2 / 2)) / 256, 256, 0, stream>>>(out_logits, out_h);
}